// BICEPLayer_22445499089477
// MI455X (gfx1250) — compile-verified
//
#include <hip/hip_runtime.h>
#include <hip/hip_bf16.h>
#include <math.h>

#define BATCH   256
#define IN_DIM  1024
#define OUT_DIM 1024
#define NPATHS  2000
#define NSTEPS  128
#define NPTOT   (BATCH * NPATHS)      // 512000 paths total
#define DT      (1.0f / 128.0f)
#define SQDT    0.08838834764831845f  // sqrt(1/128)

typedef float v2f __attribute__((ext_vector_type(2)));
typedef float v8f __attribute__((ext_vector_type(8)));

// ---------------------------------------------------------------------------
// Kernel 1: feedback drift factors a[b,g] = 1 + (sigmoid(x.W_in[g]+b_in[g]) - d)*dt
// Only the first 10 rows of W_in are ever used by the reference.
// ---------------------------------------------------------------------------
__global__ __launch_bounds__(64) void bicep_fb_kernel(
    const float* __restrict__ x, const float* __restrict__ W_in,
    const float* __restrict__ b_in, const float* __restrict__ decay,
    float* __restrict__ a_ws) {
  __shared__ float red[64];
  const int bg = blockIdx.x;            // 0..2559
  const int b = bg / 10, g = bg % 10;
  const float* xr = x + b * IN_DIM;
  const float* wr = W_in + g * IN_DIM;
  float s = 0.0f;
  for (int i = threadIdx.x; i < IN_DIM; i += 64) s = fmaf(xr[i], wr[i], s);
  red[threadIdx.x] = s;
  __syncthreads();
  for (int off = 32; off > 0; off >>= 1) {
    if (threadIdx.x < off) red[threadIdx.x] += red[threadIdx.x + off];
    __syncthreads();
  }
  if (threadIdx.x == 0) {
    float proj = red[0] + b_in[g];
    float fb = 1.0f / (1.0f + expf(-proj));
    a_ws[bg] = 1.0f + (fb - decay[0]) * DT;
  }
}

// ---------------------------------------------------------------------------
// Kernel 2: fused AR(1) path scan + per-path weighted time reductions.
// One thread per path; eps reads are fully coalesced (thread id == innermost
// path index). S_k[b,p] = sum_t Wc[t,k] * X_t  (t=0 term is zero since X_0=0).
// ---------------------------------------------------------------------------
__global__ __launch_bounds__(256) void bicep_path_kernel(
    const float* __restrict__ eps, const float* __restrict__ W_conv,
    const float* __restrict__ a_ws,
    float* __restrict__ S0, float* __restrict__ S1, float* __restrict__ S2) {
  __shared__ float wc[(NSTEPS + 1) * 3];
  for (int i = threadIdx.x; i < (NSTEPS + 1) * 3; i += 256) wc[i] = W_conv[i];
  __syncthreads();

  const int id = blockIdx.x * 256 + threadIdx.x;  // 0..511999
  const int b  = id / NPATHS;
  const int pg = id - b * NPATHS;                 // global path index in [0,2000)
  const int g  = pg / 200;                        // feedback group
  const float aa = a_ws[b * 10 + g];

  float X = 0.0f, s0 = 0.0f, s1 = 0.0f, s2 = 0.0f;
  const float* ep = eps + id;                     // eps[t,*] stride = NPTOT
  for (int t = 1; t <= NSTEPS; ++t) {
    float e = ep[(t - 1) * NPTOT];
    X  = fmaf(X, aa, SQDT * e);
    s0 = fmaf(wc[t * 3 + 0], X, s0);
    s1 = fmaf(wc[t * 3 + 1], X, s1);
    s2 = fmaf(wc[t * 3 + 2], X, s2);
  }
  S0[id] = s0; S1[id] = s1; S2[id] = s2;
}

// ---------------------------------------------------------------------------
// Kernel 3: conv neighbor combine along the path axis (zero-padded ends).
// agg[b,p] = S0[b,p-1] + S1[b,p] + S2[b,p+1] + b_conv
// ---------------------------------------------------------------------------
__global__ __launch_bounds__(256) void bicep_agg_kernel(
    const float* __restrict__ S0, const float* __restrict__ S1,
    const float* __restrict__ S2, const float* __restrict__ b_conv,
    float* __restrict__ agg) {
  const int id = blockIdx.x * 256 + threadIdx.x;
  const int pg = id % NPATHS;
  float v = S1[id] + b_conv[0];
  if (pg > 0)          v += S0[id - 1];
  if (pg < NPATHS - 1) v += S2[id + 1];
  agg[id] = v;
}

// ---------------------------------------------------------------------------
// Kernel 4: out = agg @ W_out.T + b_out  via V_WMMA_F32_16X16X4_F32.
// One wave per 16x16 output tile; K = 2000 in steps of 4.
// Lane decomposition: half = lane>>4, idx = lane&15.
//   A frag: a[v] = agg[mTile*16+idx][k + 2*half + v]
//   B frag: b[v] = W_out[nTile*16+idx][k + 2*half + v]   (B[k][n] = W_out[n][k])
//   C/D:    c[r] = out[mTile*16 + r + 8*half][nTile*16 + idx]
// ---------------------------------------------------------------------------
__global__ __launch_bounds__(256) void bicep_gemm_kernel(
    const float* __restrict__ agg, const float* __restrict__ W_out,
    const float* __restrict__ b_out, float* __restrict__ out) {
  const int lane = threadIdx.x & 31;
  const int wave = threadIdx.x >> 5;
  const int tile = blockIdx.x * 8 + wave;   // 1024 tiles = 16 (M) x 64 (N)
  const int mTile = tile >> 6;
  const int nTile = tile & 63;
  const int half = lane >> 4;
  const int idx  = lane & 15;

  const float* arow = agg   + (mTile * 16 + idx) * NPATHS + 2 * half;
  const float* brow = W_out + (nTile * 16 + idx) * NPATHS + 2 * half;

  v8f c = {};
  for (int k = 0; k < NPATHS; k += 4) {
    v2f a = *(const v2f*)(arow + k);
    v2f b = *(const v2f*)(brow + k);
    c = __builtin_amdgcn_wmma_f32_16x16x4_f32(
        /*neg_a=*/false, a, /*neg_b=*/false, b,
        /*c_mod=*/(short)0, c, /*reuse_a=*/false, /*reuse_b=*/false);
  }

  const float bb = b_out[nTile * 16 + idx];
  float* orow = out + (mTile * 16 + 8 * half) * OUT_DIM + nTile * 16 + idx;
#pragma unroll
  for (int r = 0; r < 8; ++r) orow[r * OUT_DIM] = c[r] + bb;
}

// ---------------------------------------------------------------------------
// Launcher. Inputs (setup_inputs order):
//  0:x 1:W_in 2:b_in 3:decay_rate 4:W_conv 5:b_conv 6:W_out 7:b_out 8:eps
// ---------------------------------------------------------------------------
extern "C" void kernel_launch(void* const* d_in, const int* in_sizes, int n_in,
                              void* d_out, int out_size, void* d_ws, size_t ws_size,
                              hipStream_t stream) {
  (void)in_sizes; (void)n_in; (void)out_size; (void)ws_size;
  const float* x      = (const float*)d_in[0];
  const float* W_in   = (const float*)d_in[1];
  const float* b_in   = (const float*)d_in[2];
  const float* decay  = (const float*)d_in[3];
  const float* W_conv = (const float*)d_in[4];
  const float* b_conv = (const float*)d_in[5];
  const float* W_out  = (const float*)d_in[6];
  const float* b_out  = (const float*)d_in[7];
  const float* eps    = (const float*)d_in[8];
  float* out = (float*)d_out;

  // Workspace layout (floats): a[2560] (pad to 4096) | S0 | S1 | S2 | agg
  float* ws   = (float*)d_ws;
  float* a_ws = ws;
  float* S0   = ws + 4096;
  float* S1   = S0 + NPTOT;
  float* S2   = S1 + NPTOT;
  float* agg  = S2 + NPTOT;

  bicep_fb_kernel<<<BATCH * 10, 64, 0, stream>>>(x, W_in, b_in, decay, a_ws);
  bicep_path_kernel<<<NPTOT / 256, 256, 0, stream>>>(eps, W_conv, a_ws, S0, S1, S2);
  bicep_agg_kernel<<<NPTOT / 256, 256, 0, stream>>>(S0, S1, S2, b_conv, agg);
  bicep_gemm_kernel<<<(16 * 64) / 8, 256, 0, stream>>>(agg, W_out, b_out, out);
}